// TorchNeighborListRangeSeparated_5214090297975
// MI455X (gfx1250) — compile-verified
//
#include <hip/hip_runtime.h>
#include <stdint.h>

// ---------------------------------------------------------------------------
// Problem constants (match reference)
// ---------------------------------------------------------------------------
#define NATOMS   1024
#define NSLOTS   14                      // slot 0 = center cell, 1..13 = image shifts
#define NROWS    (NSLOTS * NATOMS)       // 14336 "pair rows" (slot, i)
#define WPR      64                      // uint16 mask words per row (1024 cols / 16)
#define MAXH     32768                   // MAX_HALF_PAIRS
#define OUTPAIRS (2 * MAXH)              // 65536 bidirectional entries per cutoff
#define CUT0_SQ  16.0f                   // 4.0^2
#define CUT1_SQ  81.0f                   // 9.0^2

typedef __attribute__((ext_vector_type(2))) float v2f;
typedef __attribute__((ext_vector_type(8))) float v8f;

// ---------------------------------------------------------------------------
// Workspace layout (bytes), 256B aligned regions
// ---------------------------------------------------------------------------
static constexpr size_t alignup(size_t x) { return (x + 255) & ~((size_t)255); }

static constexpr size_t OFF_PNORM = 0;                                       // 1024 f32
static constexpr size_t OFF_POSA  = alignup(OFF_PNORM + NATOMS * 4);         // 1024 float4 (-2*p, pad 0)
static constexpr size_t OFF_QNORM = alignup(OFF_POSA  + NATOMS * 16);        // 14336 f32
static constexpr size_t OFF_QPAD  = alignup(OFF_QNORM + NROWS * 4);          // 14336 float4 (p - shift@cell)
static constexpr size_t OFF_OFFV  = alignup(OFF_QPAD  + NROWS * 16);         // 14 float4 (shift@cell)
static constexpr size_t OFF_MASK0 = alignup(OFF_OFFV  + NSLOTS * 16);        // 14336*64 u16
static constexpr size_t OFF_MASK1 = alignup(OFF_MASK0 + (size_t)NROWS * WPR * 2);
static constexpr size_t OFF_CNT0  = alignup(OFF_MASK1 + (size_t)NROWS * WPR * 2);
static constexpr size_t OFF_CNT1  = alignup(OFF_CNT0  + NROWS * 4);
static constexpr size_t OFF_BASE0 = alignup(OFF_CNT1  + NROWS * 4);
static constexpr size_t OFF_BASE1 = alignup(OFF_BASE0 + NROWS * 4);
static constexpr size_t OFF_META0 = alignup(OFF_BASE1 + NROWS * 4);          // [total, valid]
static constexpr size_t OFF_META1 = alignup(OFF_META0 + 64);
static constexpr size_t OFF_H0    = alignup(OFF_META1 + 64);                 // i,j,slot int32 x MAXH
static constexpr size_t OFF_H1    = alignup(OFF_H0    + (size_t)3 * MAXH * 4);
static constexpr size_t OFF_AB0   = alignup(OFF_H1    + (size_t)3 * MAXH * 4); // cA,cB,atomBase x1024
static constexpr size_t OFF_AB1   = alignup(OFF_AB0   + (size_t)3 * NATOMS * 4);
static constexpr size_t WS_NEED   = alignup(OFF_AB1   + (size_t)3 * NATOMS * 4);

// ---------------------------------------------------------------------------
// K0: precompute -2*p (A operand), |p|^2, q = p - shift@cell (B operand), |q|^2,
//     and per-slot cartesian offsets shift@cell.
// ---------------------------------------------------------------------------
__global__ void k_prepare(const float* __restrict__ pos,
                          const float* __restrict__ cell,
                          const float* __restrict__ shifts,
                          float*  __restrict__ pnorm,
                          float4* __restrict__ posA,
                          float*  __restrict__ qnorm,
                          float4* __restrict__ qpad,
                          float*  __restrict__ offv) {
  int idx = blockIdx.x * blockDim.x + threadIdx.x;
  if (idx >= NROWS) return;
  int slot = idx >> 10, j = idx & 1023;
  float px = pos[3 * j], py = pos[3 * j + 1], pz = pos[3 * j + 2];
  float tx = 0.f, ty = 0.f, tz = 0.f;
  if (slot > 0) {
    float sx = shifts[3 * (slot - 1)], sy = shifts[3 * (slot - 1) + 1], sz = shifts[3 * (slot - 1) + 2];
    tx = sx * cell[0] + sy * cell[3] + sz * cell[6];
    ty = sx * cell[1] + sy * cell[4] + sz * cell[7];
    tz = sx * cell[2] + sy * cell[5] + sz * cell[8];
  }
  float qx = px - tx, qy = py - ty, qz = pz - tz;
  qpad[idx]  = make_float4(qx, qy, qz, 0.f);
  qnorm[idx] = qx * qx + qy * qy + qz * qz;
  if (slot == 0) {
    posA[j]  = make_float4(-2.f * px, -2.f * py, -2.f * pz, 0.f);
    pnorm[j] = px * px + py * py + pz * pz;
    if (j < NSLOTS) {   // j doubles as slot id here
      float ox = 0.f, oy = 0.f, oz = 0.f;
      if (j > 0) {
        float sx = shifts[3 * (j - 1)], sy = shifts[3 * (j - 1) + 1], sz = shifts[3 * (j - 1) + 2];
        ox = sx * cell[0] + sy * cell[3] + sz * cell[6];
        oy = sx * cell[1] + sy * cell[4] + sz * cell[7];
        oz = sx * cell[2] + sy * cell[5] + sz * cell[8];
      }
      offv[4 * j] = ox; offv[4 * j + 1] = oy; offv[4 * j + 2] = oz; offv[4 * j + 3] = 0.f;
    }
  }
}

// ---------------------------------------------------------------------------
// K1: one wave per 16x16 distance tile. d2 = C + A*B with A=-2p (16x4),
//     B=q (4x16), C preloaded with |p_i|^2+|q_j|^2. Emits 16-bit cutoff masks.
// ---------------------------------------------------------------------------
__global__ void k_wmma(const float4* __restrict__ posA,
                       const float*  __restrict__ pnorm,
                       const float4* __restrict__ qpad,
                       const float*  __restrict__ qnorm,
                       uint16_t* __restrict__ mask0,
                       uint16_t* __restrict__ mask1) {
  const unsigned lane  = threadIdx.x & 31u;
  const unsigned gwave = blockIdx.x * (blockDim.x >> 5) + (threadIdx.x >> 5);
  const unsigned rowTile = gwave >> 6;       // 0..895  (slot*64 + i-tile)
  const unsigned ct      = gwave & 63u;      // j tile (16 cols each)
  const unsigned slot    = rowTile >> 6;
  const unsigned it      = rowTile & 63u;
  const unsigned iBase   = it << 4, jBase = ct << 4;
  const unsigned half    = lane >> 4, l15 = lane & 15u;

  // A: 16x4 f32 — lanes 0-15 rows M, VGPR0/1 = K0/K1; lanes 16-31 = K2/K3
  float4 pa = posA[iBase + l15];
  v2f a; a.x = half ? pa.z : pa.x; a.y = half ? pa.w : pa.y;
  // B: 4x16 f32 — mirrored layout (lane = column N)
  float4 qb = qpad[slot * NATOMS + jBase + l15];
  v2f b; b.x = half ? qb.z : qb.x; b.y = half ? qb.w : qb.y;

  const float qn = qnorm[slot * NATOMS + jBase + l15];
  v8f c;
#pragma unroll
  for (int r = 0; r < 8; ++r)                // C layout: VGPR r -> rows r / r+8, lane&15 -> col
    c[r] = pnorm[iBase + half * 8 + r] + qn;

  c = __builtin_amdgcn_wmma_f32_16x16x4_f32(false, a, false, b, (short)0, c, false, false);

  const unsigned jg = jBase + l15;
#pragma unroll
  for (int r = 0; r < 8; ++r) {
    unsigned ig = iBase + half * 8 + r;
    bool keep = (slot != 0) || (ig < jg);    // center cell: upper triangle only
    float d2 = c[r];
    unsigned b0 = (unsigned)__ballot(keep && (d2 < CUT0_SQ));
    unsigned b1 = (unsigned)__ballot(keep && (d2 < CUT1_SQ));
    if (lane == 0) {
      size_t rowLo = (size_t)slot * NATOMS + iBase + r;
      mask0[rowLo * WPR + ct]       = (uint16_t)(b0 & 0xFFFFu);
      mask0[(rowLo + 8) * WPR + ct] = (uint16_t)(b0 >> 16);
      mask1[rowLo * WPR + ct]       = (uint16_t)(b1 & 0xFFFFu);
      mask1[(rowLo + 8) * WPR + ct] = (uint16_t)(b1 >> 16);
    }
  }
}

// ---------------------------------------------------------------------------
// K2: per-row popcount
// ---------------------------------------------------------------------------
__global__ void k_count(const uint16_t* __restrict__ mask, uint32_t* __restrict__ cnt) {
  int row = blockIdx.x * blockDim.x + threadIdx.x;
  if (row >= NROWS) return;
  const uint32_t* w = (const uint32_t*)(mask + (size_t)row * WPR);
  unsigned s = 0;
#pragma unroll
  for (int k = 0; k < WPR / 2; ++k) s += __popc(w[k]);
  cnt[row] = s;
}

// K3: exclusive scan over rows (row order == jnp.nonzero enumeration order)
__global__ void k_scan_rows(const uint32_t* __restrict__ cnt,
                            uint32_t* __restrict__ base,
                            uint32_t* __restrict__ meta) {
  unsigned acc = 0;
  for (int r = 0; r < NROWS; ++r) { base[r] = acc; acc += cnt[r]; }
  meta[0] = acc;
  meta[1] = acc < MAXH ? acc : MAXH;   // capacity truncation
}

// Init: half-pair lists to -1
__global__ void k_init_half(int* __restrict__ h0, int* __restrict__ h1) {
  int idx = blockIdx.x * blockDim.x + threadIdx.x;
  if (idx < 3 * MAXH) { h0[idx] = -1; h1[idx] = -1; }
}

// Init: outputs — index regions -1, offset regions 0
__global__ void k_init_out(float* __restrict__ out) {
  int idx = blockIdx.x * blockDim.x + threadIdx.x;
  if (idx >= 2 * 5 * OUTPAIRS) return;
  int local = idx % (5 * OUTPAIRS);
  out[idx] = (local < 2 * OUTPAIRS) ? -1.0f : 0.0f;
}

// ---------------------------------------------------------------------------
// K4: wave per row — stable ordered emission of half pairs (i, j, slot)
// ---------------------------------------------------------------------------
__global__ void k_emit(const uint16_t* __restrict__ mask,
                       const uint32_t* __restrict__ base,
                       const uint32_t* __restrict__ meta,
                       int* __restrict__ hi, int* __restrict__ hj, int* __restrict__ hs) {
  unsigned lane = threadIdx.x & 31u;
  unsigned row  = blockIdx.x * (blockDim.x >> 5) + (threadIdx.x >> 5);
  if (row >= NROWS) return;                  // wave-uniform
  unsigned valid   = meta[1];
  unsigned running = base[row];
  int i_atom = row & 1023, slot = row >> 10;
  for (int t = 0; t < 2; ++t) {
    unsigned w    = t * 32 + lane;
    unsigned word = mask[(size_t)row * WPR + w];
    unsigned cw   = __popc(word);
    unsigned incl = cw;
    for (int d = 1; d < 32; d <<= 1) {       // wave32 inclusive prefix
      unsigned v = __shfl_up(incl, d, 32);
      if (lane >= (unsigned)d) incl += v;
    }
    unsigned pos = running + (incl - cw);
    unsigned bits = word;
    while (bits) {
      int bit = __ffs(bits) - 1;
      bits &= bits - 1;
      if (pos < valid) {
        hi[pos] = i_atom;
        hj[pos] = (int)(w * 16u + (unsigned)bit);
        hs[pos] = slot;
      }
      pos++;
    }
    running += __shfl(incl, 31, 32);
  }
}

// ---------------------------------------------------------------------------
// K5: wave per atom — counts for the counting sort (cA: appears as i, cB: as j)
// ---------------------------------------------------------------------------
__global__ void k_count_ab(const int* __restrict__ hi, const int* __restrict__ hj,
                           uint32_t* __restrict__ cA, uint32_t* __restrict__ cB) {
  unsigned lane = threadIdx.x & 31u;
  unsigned a = blockIdx.x * (blockDim.x >> 5) + (threadIdx.x >> 5);
  unsigned ca = 0, cb = 0;
  for (unsigned p = 0; p < MAXH; p += 32) {
    int i = hi[p + lane];
    int j = hj[p + lane];
    ca += __popc((unsigned)__ballot(i == (int)a));
    cb += __popc((unsigned)__ballot(j == (int)a));
  }
  if (lane == 0) { cA[a] = ca; cB[a] = cb; }
}

// K6: exclusive scan over atoms
__global__ void k_scan_atoms(const uint32_t* __restrict__ cA,
                             const uint32_t* __restrict__ cB,
                             uint32_t* __restrict__ ab) {
  unsigned acc = 0;
  for (int a = 0; a < NATOMS; ++a) { ab[a] = acc; acc += cA[a] + cB[a]; }
}

// ---------------------------------------------------------------------------
// K7: wave per atom — stable counting-sort scatter of bidirectional pairs.
//     Per atom: first-half entries (i==a, offset=-shift@cell) precede
//     second-half entries (j==a, offset=+shift@cell), each in selection order
//     — exactly jnp.argsort's stable sort of [H;S] by idx_i.
// ---------------------------------------------------------------------------
__global__ void k_scatter(const int* __restrict__ hi, const int* __restrict__ hj,
                          const int* __restrict__ hs,
                          const uint32_t* __restrict__ cA,
                          const uint32_t* __restrict__ ab,
                          const float* __restrict__ offv,
                          float* __restrict__ oi, float* __restrict__ oj,
                          float* __restrict__ oo) {
  unsigned lane = threadIdx.x & 31u;
  unsigned a = blockIdx.x * (blockDim.x >> 5) + (threadIdx.x >> 5);
  unsigned ltm = (1u << lane) - 1u;
  unsigned baseH = ab[a];
  unsigned baseS = ab[a] + cA[a];
  unsigned runA = 0, runB = 0;
  for (unsigned p = 0; p < MAXH; p += 32) {
    int i = hi[p + lane];
    int j = hj[p + lane];
    int s = hs[p + lane];
    unsigned mA = (unsigned)__ballot(i == (int)a);
    unsigned mB = (unsigned)__ballot(j == (int)a);
    if (i == (int)a) {
      unsigned pos = baseH + runA + __popc(mA & ltm);
      oi[pos] = (float)a; oj[pos] = (float)j;
      oo[3 * pos + 0] = -offv[4 * s + 0];
      oo[3 * pos + 1] = -offv[4 * s + 1];
      oo[3 * pos + 2] = -offv[4 * s + 2];
    }
    if (j == (int)a) {
      unsigned pos = baseS + runB + __popc(mB & ltm);
      oi[pos] = (float)a; oj[pos] = (float)i;
      oo[3 * pos + 0] = offv[4 * s + 0];
      oo[3 * pos + 1] = offv[4 * s + 1];
      oo[3 * pos + 2] = offv[4 * s + 2];
    }
    runA += __popc(mA);
    runB += __popc(mB);
  }
}

// ---------------------------------------------------------------------------
extern "C" void kernel_launch(void* const* d_in, const int* in_sizes, int n_in,
                              void* d_out, int out_size, void* d_ws, size_t ws_size,
                              hipStream_t stream) {
  (void)in_sizes; (void)n_in; (void)out_size;
  if (ws_size < WS_NEED) return;  // scratch too small: bail (compile-only env)

  const float* pos    = (const float*)d_in[0];
  const float* cell   = (const float*)d_in[1];
  const float* shifts = (const float*)d_in[2];
  float* out = (float*)d_out;

  char* ws = (char*)d_ws;
  float*    pnorm = (float*)   (ws + OFF_PNORM);
  float4*   posA  = (float4*)  (ws + OFF_POSA);
  float*    qnorm = (float*)   (ws + OFF_QNORM);
  float4*   qpad  = (float4*)  (ws + OFF_QPAD);
  float*    offv  = (float*)   (ws + OFF_OFFV);
  uint16_t* mask0 = (uint16_t*)(ws + OFF_MASK0);
  uint16_t* mask1 = (uint16_t*)(ws + OFF_MASK1);
  uint32_t* cnt0  = (uint32_t*)(ws + OFF_CNT0);
  uint32_t* cnt1  = (uint32_t*)(ws + OFF_CNT1);
  uint32_t* base0 = (uint32_t*)(ws + OFF_BASE0);
  uint32_t* base1 = (uint32_t*)(ws + OFF_BASE1);
  uint32_t* meta0 = (uint32_t*)(ws + OFF_META0);
  uint32_t* meta1 = (uint32_t*)(ws + OFF_META1);
  int* h0 = (int*)(ws + OFF_H0);
  int* h1 = (int*)(ws + OFF_H1);
  uint32_t* ab0 = (uint32_t*)(ws + OFF_AB0);  // [cA | cB | atomBase]
  uint32_t* ab1 = (uint32_t*)(ws + OFF_AB1);

  int* h0i = h0, *h0j = h0 + MAXH, *h0s = h0 + 2 * MAXH;
  int* h1i = h1, *h1j = h1 + MAXH, *h1s = h1 + 2 * MAXH;
  uint32_t* cA0 = ab0, *cB0 = ab0 + NATOMS, *atomB0 = ab0 + 2 * NATOMS;
  uint32_t* cA1 = ab1, *cB1 = ab1 + NATOMS, *atomB1 = ab1 + 2 * NATOMS;

  // output: i0 j0 o0 i1 j1 o1 flattened
  float* oi0 = out;
  float* oj0 = out + OUTPAIRS;
  float* oo0 = out + 2 * OUTPAIRS;
  float* oi1 = out + 5 * OUTPAIRS;
  float* oj1 = out + 5 * OUTPAIRS + OUTPAIRS;
  float* oo1 = out + 5 * OUTPAIRS + 2 * OUTPAIRS;

  k_init_out <<<(2 * 5 * OUTPAIRS) / 256, 256, 0, stream>>>(out);
  k_init_half<<<(3 * MAXH + 255) / 256, 256, 0, stream>>>(h0, h1);
  k_prepare  <<<NROWS / 256, 256, 0, stream>>>(pos, cell, shifts, pnorm, posA, qnorm, qpad, offv);

  // 896 row-tiles * 64 col-tiles = 57344 waves, 8 waves per 256-thread block
  k_wmma<<<57344 / 8, 256, 0, stream>>>(posA, pnorm, qpad, qnorm, mask0, mask1);

  k_count<<<NROWS / 256, 256, 0, stream>>>(mask0, cnt0);
  k_count<<<NROWS / 256, 256, 0, stream>>>(mask1, cnt1);
  k_scan_rows<<<1, 1, 0, stream>>>(cnt0, base0, meta0);
  k_scan_rows<<<1, 1, 0, stream>>>(cnt1, base1, meta1);

  k_emit<<<NROWS / 8, 256, 0, stream>>>(mask0, base0, meta0, h0i, h0j, h0s);
  k_emit<<<NROWS / 8, 256, 0, stream>>>(mask1, base1, meta1, h1i, h1j, h1s);

  k_count_ab<<<NATOMS / 8, 256, 0, stream>>>(h0i, h0j, cA0, cB0);
  k_count_ab<<<NATOMS / 8, 256, 0, stream>>>(h1i, h1j, cA1, cB1);
  k_scan_atoms<<<1, 1, 0, stream>>>(cA0, cB0, atomB0);
  k_scan_atoms<<<1, 1, 0, stream>>>(cA1, cB1, atomB1);

  k_scatter<<<NATOMS / 8, 256, 0, stream>>>(h0i, h0j, h0s, cA0, atomB0, offv, oi0, oj0, oo0);
  k_scatter<<<NATOMS / 8, 256, 0, stream>>>(h1i, h1j, h1s, cA1, atomB1, offv, oi1, oj1, oo1);
}